// GaussianBlurNaiveLayer_46342697124223
// MI455X (gfx1250) — compile-verified
//
#include <hip/hip_runtime.h>
#include <cstddef>

typedef __attribute__((ext_vector_type(2))) float v2f;
typedef __attribute__((ext_vector_type(8))) float v8f;

#define RAD 40
#define KL  81
#define KP  112            // padded LUT size: logical index 15 + (u - n), range [0,110]
#define KOFF 15
#define IMN 512
#define IMSZ (512*512)

// ---- reflect ('reflect'/symmetric) index, n = 512 ----
__device__ __forceinline__ int reflect512(int i) {
    int j = i & 1023;               // mod 1024 (two's complement safe)
    return (j < 512) ? j : (1023 - j);
}

// ---------------------------------------------------------------------------
// Pass 0: per-batch tables: kernel[81], radius, 3x3 channel-mix matrix
// ---------------------------------------------------------------------------
__global__ void make_tables(const float* __restrict__ sigmas,
                            const int*   __restrict__ steps,
                            float* __restrict__ kern,   // [64][81]
                            float* __restrict__ M3,     // [64][3][3]
                            int*   __restrict__ rad) {  // [64]
    int b = blockIdx.x;
    int tid = threadIdx.x;
    __shared__ float sw[KL];
    __shared__ float ssum;
    float sigma = sigmas[steps[b]];
    int r = (int)floorf(4.0f * sigma + 0.5f);
    if (tid < KL) {
        int pos = tid - RAD;
        float w = expf(-0.5f * (float)(pos * pos) / (sigma * sigma));
        if (pos < -r || pos > r) w = 0.0f;
        sw[tid] = w;
    }
    __syncthreads();
    if (tid == 0) {
        float s = 0.0f;
        for (int i = 0; i < KL; ++i) s += sw[i];
        ssum = s;
        rad[b] = r;
    }
    __syncthreads();
    if (tid < KL) kern[b * KL + tid] = sw[tid] / ssum;
    if (tid < 3) {
        // channel axis: length-3 reflect conv with the same kernel == 3x3 matrix
        float m[3] = {0.0f, 0.0f, 0.0f};
        for (int t = 0; t < KL; ++t) {
            int j = (tid + t - RAD) % 6;
            if (j < 0) j += 6;
            int cp = (j < 3) ? j : (5 - j);
            m[cp] += sw[t] / ssum;
        }
        for (int c = 0; c < 3; ++c) M3[b * 9 + tid * 3 + c] = m[c];
    }
}

// ---------------------------------------------------------------------------
// Pass 1: row (W-axis) convolution via f32 WMMA.  x -> tmp
// Block: 128 threads (4 waves). Wave tile: 16 rows x 32 output cols
// (two 16x16 WMMA tiles sharing one kernel-band B fragment).
// out[m][n] = sum_u  x_pad[m][n0+u] * K[u][n],  K[u][n] = k[u-n] (banded)
// ---------------------------------------------------------------------------
#define P1_H   16
#define P1_W   128
#define P1_PW  (P1_W + 2*RAD)   // 208 padded cols
#define P1_STR (P1_PW + 1)      // 209, avoids bank conflicts on row-indexed reads

__global__ void row_conv(const float* __restrict__ x,
                         float* __restrict__ tmp,
                         const float* __restrict__ kern,
                         const int* __restrict__ rad) {
    __shared__ float sx[P1_H][P1_STR];
    __shared__ float skp[KP];

    int bc = blockIdx.z;
    int b  = bc / 3;
    int h0 = blockIdx.y * P1_H;
    int w0 = blockIdx.x * P1_W;
    int tid = threadIdx.x;

    if (tid < KP) {
        int t = tid - KOFF;
        skp[tid] = (t >= 0 && t < KL) ? kern[b * KL + t] : 0.0f;
    }
    int r = rad[b];

    const float* xim = x + (size_t)bc * IMSZ;
    for (int i = tid; i < P1_H * P1_PW; i += 128) {
        int m = i / P1_PW, p = i - m * P1_PW;
        int col = reflect512(w0 + p - RAD);
        sx[m][p] = xim[(size_t)(h0 + m) * IMN + col];
    }
    __syncthreads();

    int lane = tid & 31;
    int wave = tid >> 5;
    int n  = lane & 15;        // output col within tile (also A-fragment row m)
    int kh = lane >> 4;        // K halves: lanes 16-31 hold K = 2,3

    int a  = 25 - r;
    int clo = (a <= 0) ? 0 : ((a + 15) >> 4);
    int chi = (55 + r) >> 4; if (chi > 5) chi = 5;

    int n0 = wave * 32;        // this wave's 32-col region; two tiles at n0, n0+16
    v8f acc0 = {0.f,0.f,0.f,0.f,0.f,0.f,0.f,0.f};
    v8f acc1 = {0.f,0.f,0.f,0.f,0.f,0.f,0.f,0.f};

    for (int c = clo; c <= chi; ++c) {
#pragma unroll
        for (int s = 0; s < 4; ++s) {
            int ks = c * 16 + s * 4 + 2 * kh;   // u base for this lane's K pair
            int bi = KOFF + ks - n;             // padded, always in range
            v2f Bf, A0, A1;
            Bf.x = skp[bi];
            Bf.y = skp[bi + 1];
            A0.x = sx[n][n0 + ks];
            A0.y = sx[n][n0 + ks + 1];
            A1.x = sx[n][n0 + 16 + ks];
            A1.y = sx[n][n0 + 16 + ks + 1];
            acc0 = __builtin_amdgcn_wmma_f32_16x16x4_f32(
                       false, A0, false, Bf, (short)0, acc0, false, false);
            acc1 = __builtin_amdgcn_wmma_f32_16x16x4_f32(
                       false, A1, false, Bf, (short)0, acc1, false, false);
        }
    }

    float* orow = tmp + (size_t)bc * IMSZ + (size_t)h0 * IMN + w0 + n0;
#pragma unroll
    for (int d = 0; d < 8; ++d) {
        int mm = d + 8 * kh;                    // C/D layout: M = d + 8*lanegroup
        orow[(size_t)mm * IMN + n]      = acc0[d];
        orow[(size_t)mm * IMN + n + 16] = acc1[d];
    }
}

// ---------------------------------------------------------------------------
// Pass 2: column (H-axis) convolution via f32 WMMA.  tmp -> out
// Block: 128 threads. Block tile: 32 rows x 64 cols. Each wave: one row-tile
// (mt) and two col-tiles sharing one kernel-band A fragment.
// out[m][n] = sum_u K'[m][u] * x_pad[row0+u][n],  K'[m][u] = k[u-m]
// ---------------------------------------------------------------------------
#define P2_H   32
#define P2_W   64
#define P2_PH  (P2_H + 2*RAD)   // 112 padded rows
#define P2_STR (P2_W + 1)       // 65, avoids bank conflicts between K halves

__global__ void col_conv(const float* __restrict__ tmp,
                         float* __restrict__ out,
                         const float* __restrict__ kern,
                         const int* __restrict__ rad) {
    __shared__ float sx[P2_PH * P2_STR];
    __shared__ float skp[KP];

    int bc = blockIdx.z;
    int b  = bc / 3;
    int h0 = blockIdx.y * P2_H;
    int w0 = blockIdx.x * P2_W;
    int tid = threadIdx.x;

    if (tid < KP) {
        int t = tid - KOFF;
        skp[tid] = (t >= 0 && t < KL) ? kern[b * KL + t] : 0.0f;
    }
    int r = rad[b];

    const float* tim = tmp + (size_t)bc * IMSZ;
    for (int i = tid; i < P2_PH * P2_W; i += 128) {
        int jr = i >> 6, col = i & 63;
        int grow = reflect512(h0 + jr - RAD);
        sx[jr * P2_STR + col] = tim[(size_t)grow * IMN + w0 + col];
    }
    __syncthreads();

    int lane = tid & 31;
    int wave = tid >> 5;
    int n  = lane & 15;        // col within tile (B), also A row m (= output row)
    int kh = lane >> 4;

    int a  = 25 - r;
    int clo = (a <= 0) ? 0 : ((a + 15) >> 4);
    int chi = (55 + r) >> 4; if (chi > 5) chi = 5;

    int tt = wave * 2;
    int mt  = tt >> 2;               // row tile (0..1), same for both of wave's tiles
    int nt0 = (tt & 3) * 16;         // first col tile base
    int nt1 = nt0 + 16;              // second col tile base

    v8f acc0 = {0.f,0.f,0.f,0.f,0.f,0.f,0.f,0.f};
    v8f acc1 = {0.f,0.f,0.f,0.f,0.f,0.f,0.f,0.f};

    for (int c = clo; c <= chi; ++c) {
#pragma unroll
        for (int s = 0; s < 4; ++s) {
            int ks = c * 16 + s * 4 + 2 * kh;
            int ai = KOFF + ks - n;                  // padded, always in range
            v2f A, B0, B1;
            A.x = skp[ai];
            A.y = skp[ai + 1];
            int rb = (mt * 16 + ks) * P2_STR;        // B[kk][n] = x_pad row
            B0.x = sx[rb           + nt0 + n];
            B0.y = sx[rb + P2_STR  + nt0 + n];
            B1.x = sx[rb           + nt1 + n];
            B1.y = sx[rb + P2_STR  + nt1 + n];
            acc0 = __builtin_amdgcn_wmma_f32_16x16x4_f32(
                       false, A, false, B0, (short)0, acc0, false, false);
            acc1 = __builtin_amdgcn_wmma_f32_16x16x4_f32(
                       false, A, false, B1, (short)0, acc1, false, false);
        }
    }

    float* obase = out + (size_t)bc * IMSZ + (size_t)(h0 + mt * 16) * IMN + w0;
#pragma unroll
    for (int d = 0; d < 8; ++d) {
        int mm = d + 8 * kh;
        obase[(size_t)mm * IMN + nt0 + n] = acc0[d];
        obase[(size_t)mm * IMN + nt1 + n] = acc1[d];
    }
}

// ---------------------------------------------------------------------------
// Pass 3: in-place 3x3 channel mix on d_out (one thread owns all 3 channels)
// ---------------------------------------------------------------------------
__global__ void chan_mix(float* __restrict__ out, const float* __restrict__ M3) {
    size_t idx = (size_t)blockIdx.x * 256 + threadIdx.x;   // over 64*512*512
    size_t b = idx >> 18;
    size_t px = idx & (size_t)(IMSZ - 1);
    const float* M = M3 + b * 9;
    size_t base = b * 3 * (size_t)IMSZ + px;
    float v0 = out[base];
    float v1 = out[base + IMSZ];
    float v2 = out[base + 2 * IMSZ];
    out[base]            = M[0]*v0 + M[1]*v1 + M[2]*v2;
    out[base + IMSZ]     = M[3]*v0 + M[4]*v1 + M[5]*v2;
    out[base + 2*IMSZ]   = M[6]*v0 + M[7]*v1 + M[8]*v2;
}

// ---------------------------------------------------------------------------
extern "C" void kernel_launch(void* const* d_in, const int* in_sizes, int n_in,
                              void* d_out, int out_size, void* d_ws, size_t ws_size,
                              hipStream_t stream) {
    const float* x     = (const float*)d_in[0];
    const float* sig   = (const float*)d_in[1];
    const int*   steps = (const int*)d_in[2];
    float* out = (float*)d_out;

    char* ws = (char*)d_ws;
    float* kern = (float*)ws;                       // 64*81 floats
    float* M3   = (float*)(ws + 64 * KL * 4);       // 64*9 floats
    int*   rad  = (int*)(ws + 64 * KL * 4 + 64 * 9 * 4);
    float* tmp  = (float*)(ws + 32768);             // 64*3*512*512 floats

    make_tables<<<64, 128, 0, stream>>>(sig, steps, kern, M3, rad);

    dim3 g1(IMN / P1_W, IMN / P1_H, 64 * 3);        // (4, 32, 192)
    row_conv<<<g1, 128, 0, stream>>>(x, tmp, kern, rad);

    dim3 g2(IMN / P2_W, IMN / P2_H, 64 * 3);        // (8, 16, 192)
    col_conv<<<g2, 128, 0, stream>>>(tmp, out, kern, rad);

    chan_mix<<<(64 * IMSZ) / 256, 256, 0, stream>>>(out, M3);
}